// StructuralViolationLoss_481036337741
// MI455X (gfx1250) — compile-verified
//
#include <hip/hip_runtime.h>
#include <hip/hip_bf16.h>

typedef __attribute__((ext_vector_type(2))) float v2f;
typedef __attribute__((ext_vector_type(8))) float v8f;

#define EPSF       1e-8f
#define IDEAL_CA   3.8f
#define BOND_TOLF  0.4f
#define CLASH_TOLF 1.5f
#define CLASH_SQ   (CLASH_TOLF * CLASH_TOLF)   // 2.25

constexpr int LCONST = 2048;
constexpr int WAVES_PER_BLOCK = 8;             // 256 threads, wave32
constexpr int CHUNKS = 4;                      // column-tile chunks per row tile

__device__ __forceinline__ float wave_reduce_sum(float v) {
    #pragma unroll
    for (int off = 16; off > 0; off >>= 1)
        v += __shfl_down(v, off, 32);
    return v;
}

// ---------------------------------------------------------------------------
// Kernel 1: bond term over B*(L-1) consecutive pairs. Each wave writes one
// deterministic (num, den) partial to the workspace.
// ---------------------------------------------------------------------------
__global__ void svl_bond_kernel(const float* __restrict__ pos,
                                const float* __restrict__ mask,
                                float* __restrict__ bondP, int B, int L) {
    int idx = blockIdx.x * blockDim.x + threadIdx.x;
    int total = B * (L - 1);
    float num = 0.0f, den = 0.0f;
    if (idx < total) {
        int b = idx / (L - 1);
        int t = idx % (L - 1);
        const float* p0 = pos + (size_t)(b * L + t) * 3;
        float dx = p0[3] - p0[0];
        float dy = p0[4] - p0[1];
        float dz = p0[5] - p0[2];
        float dist = sqrtf(dx * dx + dy * dy + dz * dz + EPSF);
        float viol = fmaxf(fabsf(dist - IDEAL_CA) - BOND_TOLF, 0.0f);
        float bm = mask[b * L + t] * mask[b * L + t + 1];
        num = viol * bm;
        den = bm;
    }
    num = wave_reduce_sum(num);
    den = wave_reduce_sum(den);
    if ((threadIdx.x & 31) == 0) {
        int w = idx >> 5;
        bondP[2 * w + 0] = num;
        bondP[2 * w + 1] = den;
    }
}

// ---------------------------------------------------------------------------
// Kernel 2: clash term. Block = 8 waves sharing one batch; LDS holds the
// batch's positions pre-padded as (x,y)(z,0) float2 pairs (one aligned
// ds_load_b64 per WMMA fragment half) plus packed (sqnorm, mask). Each wave
// owns a (16-row tile, column chunk) pair and sweeps 32 column tiles with
// f32 WMMA: ||xi-xj||^2 = ||xi||^2 + ||xj||^2 - 2 (X_i . X_j).
// Wave-uniform indices are pinned to SGPRs via readfirstlane so all loop
// control is scalar (EXEC stays all-ones for WMMA). Far tiles (|ti-tj|>=2)
// have guaranteed seq-sep > 2: denominator is one FMA and the sqrt path
// hides behind a wave-uniform ballot on a min-tree of d^2.
// ---------------------------------------------------------------------------
__global__ void __launch_bounds__(256)
svl_clash_kernel(const float* __restrict__ pos,
                 const float* __restrict__ mask,
                 float* __restrict__ clashP, int B, int L) {
    __shared__ float2 sPosP[LCONST * 2];  // 32 KB: point j -> {(x,y),(z,0)}
    __shared__ float2 sSM[LCONST];        // 16 KB: (sqnorm, mask)

    const int lane = threadIdx.x & 31;
    // Wave-uniform job id pinned to an SGPR.
    const int wflat = __builtin_amdgcn_readfirstlane(
        blockIdx.x * WAVES_PER_BLOCK + (threadIdx.x >> 5));
    const int tilesPerB = L >> 4;                  // 128
    const int c   = wflat % CHUNKS;
    const int tiG = wflat / CHUNKS;
    const int ti  = tiG % tilesPerB;
    const int b   = tiG / tilesPerB;
    const int ibase = ti << 4;
    const int h    = lane >> 4;
    const int nloc = lane & 15;

    const float* P = pos  + (size_t)b * L * 3;
    const float* M = mask + (size_t)b * L;

    // Single-pass cooperative staging: pad positions, fuse sq-norm compute.
    for (int idx = threadIdx.x; idx < L; idx += blockDim.x) {
        float x = P[3 * idx + 0];
        float y = P[3 * idx + 1];
        float z = P[3 * idx + 2];
        sPosP[2 * idx + 0] = make_float2(x, y);
        sPosP[2 * idx + 1] = make_float2(z, 0.0f);
        sSM[idx] = make_float2(x * x + y * y + z * z, M[idx]);
    }
    __syncthreads();

    // A fragment (16x4 f32, K padded 3->4): lane holds X[ibase+nloc][k],
    // VGPR v -> k = v + 2*h; the k==3 pad is a real zero in LDS.
    v2f afrag;
    {
        float2 at = sPosP[2 * (ibase + nloc) + h];
        afrag.x = at.x;
        afrag.y = at.y;
    }

    // Row-side norms and masks for this lane's 8 output rows: m = r + 8*h.
    float sqi[8], mi[8];
    #pragma unroll
    for (int r = 0; r < 8; ++r) {
        float2 sm = sSM[ibase + r + 8 * h];
        sqi[r] = sm.x;
        mi[r]  = sm.y;
    }
    float miSum = 0.0f;
    #pragma unroll
    for (int r = 0; r < 8; ++r) miSum += mi[r];

    float num = 0.0f, den = 0.0f;

    auto far_tile = [&](int tj) {
        int j = (tj << 4) + nloc;
        float2 bt = sPosP[2 * j + h];
        v2f bfrag; bfrag.x = bt.x; bfrag.y = bt.y;
        float2 sm = sSM[j];

        v8f g = {0.0f, 0.0f, 0.0f, 0.0f, 0.0f, 0.0f, 0.0f, 0.0f};
        g = __builtin_amdgcn_wmma_f32_16x16x4_f32(
                false, afrag, false, bfrag, (short)0, g, false, false);

        float d2[8];
        #pragma unroll
        for (int r = 0; r < 8; ++r)
            d2[r] = (sqi[r] + sm.x) - 2.0f * g[r];

        den += miSum * sm.y;

        float mn = fminf(fminf(fminf(d2[0], d2[1]), fminf(d2[2], d2[3])),
                         fminf(fminf(d2[4], d2[5]), fminf(d2[6], d2[7])));
        if (__ballot(mn < CLASH_SQ) != 0ull) {
            #pragma unroll
            for (int r = 0; r < 8; ++r) {
                float dist = sqrtf(fmaxf(d2[r], 0.0f) + EPSF);
                num += fmaxf(CLASH_TOLF - dist, 0.0f) * (mi[r] * sm.y);
            }
        }
    };

    auto near_tile = [&](int tj) {
        int j = (tj << 4) + nloc;
        float2 bt = sPosP[2 * j + h];
        v2f bfrag; bfrag.x = bt.x; bfrag.y = bt.y;
        float2 sm = sSM[j];

        v8f g = {0.0f, 0.0f, 0.0f, 0.0f, 0.0f, 0.0f, 0.0f, 0.0f};
        g = __builtin_amdgcn_wmma_f32_16x16x4_f32(
                false, afrag, false, bfrag, (short)0, g, false, false);

        #pragma unroll
        for (int r = 0; r < 8; ++r) {
            int i = ibase + r + 8 * h;
            int sep = i - j; sep = (sep < 0) ? -sep : sep;
            float pm = (sep > 2) ? (mi[r] * sm.y) : 0.0f;
            float d2 = (sqi[r] + sm.x) - 2.0f * g[r];
            float dist = sqrtf(fmaxf(d2, 0.0f) + EPSF);
            num += fmaxf(CLASH_TOLF - dist, 0.0f) * pm;
            den += pm;
        }
    };

    // This wave's column-tile window [lo, hi) intersected with the
    // near-diagonal band [nearLo, nearHi1).
    const int chunkLen = tilesPerB / CHUNKS;     // 32
    const int lo = c * chunkLen;
    const int hi = lo + chunkLen;
    const int nearLo  = (ti - 1 < 0) ? 0 : ti - 1;
    const int nearHi1 = (ti + 2 > tilesPerB) ? tilesPerB : ti + 2;

    const int fa1 = (hi < nearLo) ? hi : nearLo;            // far below band
    const int nr0 = (lo > nearLo) ? lo : nearLo;            // near band
    const int nr1 = (hi < nearHi1) ? hi : nearHi1;
    const int fb0 = (lo > nearHi1) ? lo : nearHi1;          // far above band

    #pragma unroll 2
    for (int tj = lo; tj < fa1; ++tj) far_tile(tj);
    for (int tj = nr0; tj < nr1; ++tj) near_tile(tj);
    #pragma unroll 2
    for (int tj = fb0; tj < hi; ++tj) far_tile(tj);

    num = wave_reduce_sum(num);
    den = wave_reduce_sum(den);
    if (lane == 0) {
        clashP[2 * wflat + 0] = num;
        clashP[2 * wflat + 1] = den;
    }
}

// ---------------------------------------------------------------------------
// Kernel 3: deterministic reduction of wave partials + final combine.
// ---------------------------------------------------------------------------
__global__ void svl_finalize_kernel(const float* __restrict__ bondP, int nBond,
                                    const float* __restrict__ clashP, int nClash,
                                    float* __restrict__ out) {
    __shared__ float sh[256][4];
    int t = threadIdx.x;
    float bn = 0.0f, bd = 0.0f, cn = 0.0f, cd = 0.0f;
    for (int i = t; i < nBond; i += 256) {
        bn += bondP[2 * i + 0];
        bd += bondP[2 * i + 1];
    }
    for (int i = t; i < nClash; i += 256) {
        cn += clashP[2 * i + 0];
        cd += clashP[2 * i + 1];
    }
    sh[t][0] = bn; sh[t][1] = bd; sh[t][2] = cn; sh[t][3] = cd;
    __syncthreads();
    if (t == 0) {
        float a = 0.0f, b2 = 0.0f, c = 0.0f, d = 0.0f;
        for (int i = 0; i < 256; ++i) {
            a += sh[i][0]; b2 += sh[i][1]; c += sh[i][2]; d += sh[i][3];
        }
        float bond  = a / (b2 + EPSF);
        float clash = c / (d + EPSF);
        out[0] = bond;
        out[1] = clash;
        out[2] = bond + clash;
    }
}

extern "C" void kernel_launch(void* const* d_in, const int* in_sizes, int n_in,
                              void* d_out, int out_size, void* d_ws, size_t ws_size,
                              hipStream_t stream) {
    (void)n_in; (void)out_size; (void)ws_size;
    const float* pos  = (const float*)d_in[0];   // [B, L, 3] f32
    const float* mask = (const float*)d_in[1];   // [B, L]    f32
    float* out = (float*)d_out;                  // [3]       f32

    const int L = LCONST;
    const int B = in_sizes[1] / L;               // 8

    const int bondBlocks = (B * (L - 1) + 255) / 256;        // 64
    const int nBondWaves = bondBlocks * (256 / 32);          // 512
    const int tilesPerB  = L / 16;                           // 128
    const int nClashWaves = B * tilesPerB * CHUNKS;          // 4096
    const int clashBlocks = nClashWaves / WAVES_PER_BLOCK;   // 512

    float* bondP  = (float*)d_ws;                // 2*nBondWaves floats
    float* clashP = bondP + 2 * nBondWaves;      // 2*nClashWaves floats

    svl_bond_kernel<<<bondBlocks, 256, 0, stream>>>(pos, mask, bondP, B, L);
    svl_clash_kernel<<<clashBlocks, 256, 0, stream>>>(pos, mask, clashP, B, L);
    svl_finalize_kernel<<<1, 256, 0, stream>>>(bondP, nBondWaves,
                                               clashP, nClashWaves, out);
}